// GNN_40157944218343
// MI455X (gfx1250) — compile-verified
//
#include <hip/hip_runtime.h>
#include <hip/hip_bf16.h>

typedef __attribute__((ext_vector_type(2))) float v2f;
typedef __attribute__((ext_vector_type(8))) float v8f;
typedef __attribute__((ext_vector_type(2))) long long v2ll;

#define GNN_IN_DIM 17

// ---------------------------------------------------------------------------
// Kernel 1: zero the segment-sum accumulators (agg, cnt).
// ---------------------------------------------------------------------------
__global__ void gnn_zero_kernel(float* __restrict__ agg, float* __restrict__ cnt,
                                int n) {
  int i = blockIdx.x * blockDim.x + threadIdx.x;
  if (i < n) {
    agg[i] = 0.0f;
    cnt[i] = 0.0f;
  }
}

// ---------------------------------------------------------------------------
// Kernel 2: node projections p_l = x.w_l, p_r = x.w_r via V_WMMA_F32_16X16X4_F32
// (exact fp32 — no precision loss vs the fp32 reference).
//
// One wave handles a 16-node tile. Per ISA layouts (CDNA5 ISA):
//   A (16x4 f32):  lane 0-15 -> M = lane, lane 16-31 -> M = lane-16;
//                  K = 2*half + vgpr_j               (half = lane>>4)
//   B (4x16 f32):  lane lid -> column N = lid;  K = 2*half + vgpr_j
//   C/D (16x16):   lane lid -> column N = lid;  element r -> row M = r + 8*half
// Column 0 of B carries w_l, column 1 carries w_r, rest zero.
//
// All loads are UNCONDITIONAL with always-valid addresses; zero-padding is done
// with value selects (v_cndmask), never predicated loads — avoids the
// saveexec/branch serialization seen in round 1. Chunks k0=0..12 have
// kb+1 = k0+2*half+1 <= 15 < 17 for every lane, so they need no masking at
// all; only the final chunk (K=16..19) masks values.
// ---------------------------------------------------------------------------
__global__ __launch_bounds__(256) void gnn_proj_wmma_kernel(
    const float* __restrict__ x, const float* __restrict__ w_l,
    const float* __restrict__ w_r, float* __restrict__ p_l,
    float* __restrict__ p_r, int n_nodes) {
  const int lane = threadIdx.x & 31;
  const int half = lane >> 4;   // 0 or 1
  const int lid  = lane & 15;
  const int wave = (int)((blockIdx.x * (unsigned)blockDim.x + threadIdx.x) >> 5);
  const int row0 = wave * 16;
  if (row0 >= n_nodes) return;  // wave-uniform: EXEC stays all-ones for WMMA

  // Row this lane supplies for the A operand (clamped for safety; N%16==0 in
  // practice so the clamp is a no-op).
  int arow = row0 + lid;
  if (arow >= n_nodes) arow = n_nodes - 1;
  const float* xrow = x + (size_t)arow * GNN_IN_DIM;
  const float* wv = (lid == 0) ? w_l : w_r;  // valid base for every lane
  const bool bcol = (lid < 2);               // B columns 0/1 carry weights

  v8f c = {0.f, 0.f, 0.f, 0.f, 0.f, 0.f, 0.f, 0.f};

#pragma unroll
  for (int k0 = 0; k0 < 16; k0 += 4) {
    const int kb = k0 + 2 * half;   // <= 15 for all lanes: fully in-bounds
    v2f a, b;
    a.x = xrow[kb + 0];
    a.y = xrow[kb + 1];
    const float bx = wv[kb + 0];    // unconditional load, value-masked below
    const float by = wv[kb + 1];
    b.x = bcol ? bx : 0.0f;
    b.y = bcol ? by : 0.0f;
    // (neg_a, A, neg_b, B, c_mod, C, reuse_a, reuse_b)
    c = __builtin_amdgcn_wmma_f32_16x16x4_f32(false, a, false, b, (short)0, c,
                                              false, false);
  }
  {  // Final chunk: K = 16..19; only K=16 is real, rest zero-padded.
    v2f a, b;
    const float ax = xrow[16];      // always in-bounds
    const float bx = wv[16];
    a.x = (half == 0) ? ax : 0.0f;
    a.y = 0.0f;
    b.x = (bcol && half == 0) ? bx : 0.0f;
    b.y = 0.0f;
    c = __builtin_amdgcn_wmma_f32_16x16x4_f32(false, a, false, b, (short)0, c,
                                              false, false);
  }

  // Write out: column 0 (lanes 0 & 16) -> p_l, column 1 (lanes 1 & 17) -> p_r.
  if (lid == 0) {
#pragma unroll
    for (int r = 0; r < 8; ++r) {
      int m = row0 + 8 * half + r;
      if (m < n_nodes) p_l[m] = c[r];
    }
  } else if (lid == 1) {
#pragma unroll
    for (int r = 0; r < 8; ++r) {
      int m = row0 + 8 * half + r;
      if (m < n_nodes) p_r[m] = c[r];
    }
  }
}

// ---------------------------------------------------------------------------
// Kernel 3: edge scatter — the roofline-setting phase.
// Streams 256 MB of int64 indices ONCE with non-temporal b128 loads (no reuse;
// keeps the hot 6 MB of p_l/agg/cnt resident in the 192 MB L2), gathers
// p_l[src] (L2-resident, regular temporal policy) and does fp32 atomics into
// L2-resident agg/cnt. 2 edges per thread -> 16B-aligned vector index loads.
// ---------------------------------------------------------------------------
__global__ void gnn_edge_kernel(const long long* __restrict__ ei,  // [2, E]
                                const float* __restrict__ p_l,
                                float* __restrict__ agg,
                                float* __restrict__ cnt, int n_edges) {
  const int t = blockIdx.x * blockDim.x + threadIdx.x;
  const int e0 = t * 2;
  if (e0 >= n_edges) return;
  if (e0 + 1 < n_edges) {
    // src pair and dst pair: 16-byte aligned (e0 even, rows are E*8B apart
    // with E even).
    const v2ll sp = __builtin_nontemporal_load((const v2ll*)ei + t);
    const v2ll dp = __builtin_nontemporal_load((const v2ll*)(ei + n_edges) + t);
    const int s0 = (int)sp.x, s1 = (int)sp.y;
    const int d0 = (int)dp.x, d1 = (int)dp.y;
    atomicAdd(&agg[d0], p_l[s0]);
    atomicAdd(&cnt[d0], 1.0f);
    atomicAdd(&agg[d1], p_l[s1]);
    atomicAdd(&cnt[d1], 1.0f);
  } else {  // odd tail edge
    const int s0 = (int)ei[e0];
    const int d0 = (int)ei[(size_t)n_edges + e0];
    atomicAdd(&agg[d0], p_l[s0]);
    atomicAdd(&cnt[d0], 1.0f);
  }
}

// ---------------------------------------------------------------------------
// Kernel 4: epilogue. mean -> + b_l + p_r -> ELU -> out linear.
// ---------------------------------------------------------------------------
__global__ void gnn_final_kernel(const float* __restrict__ agg,
                                 const float* __restrict__ cnt,
                                 const float* __restrict__ p_r,
                                 const float* __restrict__ b_l,
                                 const float* __restrict__ w_o,
                                 const float* __restrict__ b_o,
                                 float* __restrict__ out, int n) {
  int i = blockIdx.x * blockDim.x + threadIdx.x;
  if (i >= n) return;
  const float mean = agg[i] / fmaxf(cnt[i], 1.0f);
  float h = mean + b_l[0] + p_r[i];
  h = (h > 0.0f) ? h : (expf(h) - 1.0f);  // ELU, alpha = 1
  out[i] = h * w_o[0] + b_o[0];
}

// ---------------------------------------------------------------------------
// Launch. Inputs (setup_inputs order): x, edge_index, edge_weight(unused),
// w_l, b_l, w_r, w_o, b_o. Workspace: p_l | p_r | agg | cnt (4*N floats = 8MB).
// ---------------------------------------------------------------------------
extern "C" void kernel_launch(void* const* d_in, const int* in_sizes, int n_in,
                              void* d_out, int out_size, void* d_ws,
                              size_t ws_size, hipStream_t stream) {
  const float* x      = (const float*)d_in[0];
  const long long* ei = (const long long*)d_in[1];  // int64 [2, E]
  const float* w_l    = (const float*)d_in[3];
  const float* b_l    = (const float*)d_in[4];
  const float* w_r    = (const float*)d_in[5];
  const float* w_o    = (const float*)d_in[6];
  const float* b_o    = (const float*)d_in[7];
  float* out = (float*)d_out;

  const int n_nodes = in_sizes[0] / GNN_IN_DIM;  // 500000
  const int n_edges = in_sizes[2];               // 16000000 (edge_weight len)

  float* p_l = (float*)d_ws;
  float* p_r = p_l + n_nodes;
  float* agg = p_r + n_nodes;
  float* cnt = agg + n_nodes;

  const int B = 256;
  const int node_blocks = (n_nodes + B - 1) / B;
  // One wave (32 threads) per 16-node tile -> 2 threads per node.
  const int proj_blocks = (n_nodes * 2 + B - 1) / B;
  // 2 edges per thread.
  const int edge_blocks = (n_edges / 2 + B - 1) / B;

  gnn_zero_kernel<<<node_blocks, B, 0, stream>>>(agg, cnt, n_nodes);
  gnn_proj_wmma_kernel<<<proj_blocks, B, 0, stream>>>(x, w_l, w_r, p_l, p_r,
                                                      n_nodes);
  gnn_edge_kernel<<<edge_blocks, B, 0, stream>>>(ei, p_l, agg, cnt, n_edges);
  gnn_final_kernel<<<node_blocks, B, 0, stream>>>(agg, cnt, p_r, b_l, w_o, b_o,
                                                  out, n_nodes);
}